// AttentionHiddenNet_85916525789414
// MI455X (gfx1250) — compile-verified
//
#include <hip/hip_runtime.h>
#include <hip/hip_bf16.h>

typedef __attribute__((ext_vector_type(2))) float v2f;
typedef __attribute__((ext_vector_type(4))) float v4f;
typedef __attribute__((ext_vector_type(8))) float v8f;

#define HDIM 64
#define PITCH 68          // 64 + 4 floats: staggers LDS banks row-to-row (row base stays 16B-aligned)
#define NTHREADS 512      // 16 wave32 -> 4x4 tile grid per phase

__global__ __launch_bounds__(NTHREADS)
void varlen_attn_wmma_kernel(const float* __restrict__ h,
                             const int* __restrict__ sse,
                             float* __restrict__ out)
{
    __shared__ float Hp[64 * PITCH];   // zero-padded sequence tile, row-major
    __shared__ float S [64 * PITCH];   // scores -> softmax probs (in place)

    const int s     = blockIdx.x;
    const int start = sse[2 * s];
    const int len   = sse[2 * s + 1] - start;
    const int nt    = (len + 15) >> 4;   // 16-tiles per dim (1..4)
    const int Lp    = nt << 4;           // padded length

    const int tid  = threadIdx.x;
    const int lane = tid & 31;
    const int wave = tid >> 5;           // 0..15
    const int ti   = wave >> 2;          // tile row 0..3
    const int tj   = wave & 3;           // tile col 0..3
    const int m    = lane & 15;          // index within tile (M for A, N for B/C/D)
    const int half = lane >> 4;          // 0 or 1 (selects K pair / +8 rows of C)

    // ---- load Hp as float4 (GLOBAL_LOAD_B128 / DS_STORE_B128), zero-pad rows >= len ----
    const float* gsrc = h + (size_t)start * HDIM;
    #pragma unroll
    for (int i = 0; i < 2; ++i) {
        int e4 = tid + i * NTHREADS;     // float4 index 0..1023
        int r  = e4 >> 4;                // 16 float4 per 64-float row
        int c4 = e4 & 15;
        v4f v = {};
        if (r < len) v = *(const v4f*)(gsrc + r * HDIM + c4 * 4);
        *(v4f*)&Hp[r * PITCH + c4 * 4] = v;
    }
    __syncthreads();

    // ---- phase 1: S = Hp * Hp^T, tile (ti,tj), K = HDIM = 64 ----
    if (ti < nt && tj < nt) {            // wave-uniform: EXEC all-1s for WMMA
        v8f acc = {};
        const float* arow = &Hp[(ti * 16 + m) * PITCH];
        const float* brow = &Hp[(tj * 16 + m) * PITCH];
        #pragma unroll
        for (int k0 = 0; k0 < HDIM; k0 += 4) {
            const int kb = k0 + half * 2;
            v2f a; a.x = arow[kb]; a.y = arow[kb + 1];
            v2f b; b.x = brow[kb]; b.y = brow[kb + 1];
            acc = __builtin_amdgcn_wmma_f32_16x16x4_f32(
                      false, a, false, b, (short)0, acc, false, false);
        }
        const int  col      = tj * 16 + m;
        const bool colvalid = (col < len);
        #pragma unroll
        for (int v = 0; v < 8; ++v) {
            int row = ti * 16 + v + half * 8;
            S[row * PITCH + col] = colvalid ? acc[v] : -1e30f;
        }
    }
    __syncthreads();

    // ---- phase 2: row softmax over the Lp computed columns ----
    if (tid < Lp) {
        float* row = &S[tid * PITCH];
        float mx = -1e30f;
        for (int c = 0; c < Lp; ++c) mx = fmaxf(mx, row[c]);
        float sum = 0.0f;
        for (int c = 0; c < Lp; ++c) {
            float e = __expf(row[c] - mx);
            row[c] = e;
            sum += e;
        }
        float inv = 1.0f / sum;
        for (int c = 0; c < Lp; ++c) row[c] *= inv;
    }
    __syncthreads();

    // ---- phase 3: ctx = P * Hp, tile grid nt x 4, K = Lp (nt chunks of 16) ----
    if (ti < nt) {                       // all 4 column tiles of HDIM active
        v8f acc = {};
        const float* prow = &S[(ti * 16 + m) * PITCH];
        const int    col  = tj * 16 + m;
        for (int kt = 0; kt < nt; ++kt) {
            const int kbase = kt * 16;
            #pragma unroll
            for (int kk = 0; kk < 16; kk += 4) {
                const int kb = kbase + kk + half * 2;
                v2f a; a.x = prow[kb]; a.y = prow[kb + 1];
                v2f b; b.x = Hp[kb * PITCH + col];
                       b.y = Hp[(kb + 1) * PITCH + col];
                acc = __builtin_amdgcn_wmma_f32_16x16x4_f32(
                          false, a, false, b, (short)0, acc, false, false);
            }
        }
        float* gdst = out + (size_t)start * HDIM;
        #pragma unroll
        for (int v = 0; v < 8; ++v) {
            int row = ti * 16 + v + half * 8;
            if (row < len) gdst[row * HDIM + col] = acc[v];
        }
    }
}

extern "C" void kernel_launch(void* const* d_in, const int* in_sizes, int n_in,
                              void* d_out, int out_size, void* d_ws, size_t ws_size,
                              hipStream_t stream) {
    const float* h   = (const float*)d_in[0];
    const int*   sse = (const int*)d_in[1];
    float*       out = (float*)d_out;
    const int n_seqs = in_sizes[1] / 2;   // (4096, 2) int pairs

    varlen_attn_wmma_kernel<<<n_seqs, NTHREADS, 0, stream>>>(h, sse, out);
    (void)d_ws; (void)ws_size; (void)n_in; (void)out_size;
}